// QuantumFraudLayer_65481071408338
// MI455X (gfx1250) — compile-verified
//
#include <hip/hip_runtime.h>
#include <hip/hip_bf16.h>

// out[i] = cosf(in[i]) elementwise over BATCH*2 f32 values.
// (Reference's 2-qubit statevector sim collapses analytically to
//  [<Z0>,<Z1>] = [cos(r1), cos(r2)] -- RZZ phase is killed by |psi|^2.)
//
// Memory-bound streaming: 128 MiB total traffic / 23.3 TB/s ~= 5.8 us floor.
// 128-bit NT loads/stores, 8 float4 per thread batched for latency hiding,
// v_cos_f32 (TRANS) for the math. Fast path is unpredicated to avoid
// exec-mask juggling and VGPR zero-init.

typedef float v4f __attribute__((ext_vector_type(4)));

#define QF_BLOCK 256
#define QF_ITER  8

__global__ __launch_bounds__(QF_BLOCK, 1)
void QuantumFraud_cos_stream_kernel(const v4f* __restrict__ in4,
                                    v4f* __restrict__ out4,
                                    const float* __restrict__ in1,
                                    float* __restrict__ out1,
                                    int n4, int n) {
    const int tid = blockIdx.x * QF_BLOCK + threadIdx.x;
    const int nthreads = gridDim.x * QF_BLOCK;

    if (tid + (QF_ITER - 1) * nthreads < n4) {
        // ---- fast path: whole batch in range, no predication ----
        v4f v[QF_ITER];
#pragma unroll
        for (int k = 0; k < QF_ITER; ++k) {
            v[k] = __builtin_nontemporal_load(&in4[tid + k * nthreads]); // b128 NT
        }
#pragma unroll
        for (int k = 0; k < QF_ITER; ++k) {
            v4f r;
            r.x = __cosf(v[k].x);   // v_mul_f32 (1/2pi) + v_cos_f32
            r.y = __cosf(v[k].y);
            r.z = __cosf(v[k].z);
            r.w = __cosf(v[k].w);
            __builtin_nontemporal_store(r, &out4[tid + k * nthreads]);   // b128 NT
        }
    } else {
        // ---- edge threads: guarded grid-stride over remaining float4s ----
        for (int i = tid; i < n4; i += nthreads) {
            v4f v = __builtin_nontemporal_load(&in4[i]);
            v4f r;
            r.x = __cosf(v.x);
            r.y = __cosf(v.y);
            r.z = __cosf(v.z);
            r.w = __cosf(v.w);
            __builtin_nontemporal_store(r, &out4[i]);
        }
    }

    // ---- scalar tail (n % 4 != 0; unused for BATCH=8M but safe) ----
    const int tail_start = n4 * 4;
    for (int i = tail_start + tid; i < n; i += nthreads) {
        out1[i] = __cosf(in1[i]);
    }
}

extern "C" void kernel_launch(void* const* d_in, const int* in_sizes, int n_in,
                              void* d_out, int out_size, void* d_ws, size_t ws_size,
                              hipStream_t stream) {
    (void)n_in; (void)d_ws; (void)ws_size;

    const float* in = (const float*)d_in[0];
    float* out = (float*)d_out;

    const int n  = out_size;          // == in_sizes[0] == BATCH*2
    const int n4 = n >> 2;            // float4 elements

    // Exact-cover geometry: each thread handles QF_ITER float4s.
    const long long threads_needed = ((long long)n4 + QF_ITER - 1) / QF_ITER;
    int grid = (int)((threads_needed + QF_BLOCK - 1) / QF_BLOCK);
    if (grid < 1) grid = 1;

    QuantumFraud_cos_stream_kernel<<<grid, QF_BLOCK, 0, stream>>>(
        (const v4f*)in, (v4f*)out, in, out, n4, n);
}